// PositionalEncodingAngle_80590766342356
// MI455X (gfx1250) — compile-verified
//
#include <hip/hip_runtime.h>

typedef __attribute__((ext_vector_type(2))) float v2f;
typedef __attribute__((ext_vector_type(8))) float v8f;

#define D_MODEL   1024
#define RES_CLAMP 45.0f
// log2(45)
#define LOG2_RES  5.491853096329675f
// 1/(2*pi): v_sin_f32 computes sin(2*pi*t), so fold the normalization into the table
#define INV_2PI   0.15915494309189535f

// Each wave handles a 16-position group across all 1024 columns as 64 WMMA
// tiles of 16x16. Block = 256 threads = 8 waves = 128 positions per block.
__global__ __launch_bounds__(256) void posenc_angle_wmma_kernel(
    const float* __restrict__ x, float* __restrict__ out) {
  // LDS: for each of 64 j-tiles, per-lane float2 {invden_row0, invden_row1},
  // pre-scaled by 1/(2*pi); lanes 16..31 hold zeros (they feed K=2,3 of B).
  __shared__ float ldsB[64 * 32 * 2];  // 16 KB

  const int tid  = threadIdx.x;
  const int lane = tid & 31;
  const int wave = tid >> 5;

  // ---- Build the B-operand table (amortized over 128*1024 outputs) ----
  const float escale = -(LOG2_RES / 512.0f);  // exponent per floor(e/2) step
#pragma unroll
  for (int k = 0; k < 8; ++k) {
    int idx = tid + k * 256;          // 0..2047 -> (jtile, lane) slots
    int jt  = idx >> 5;
    int ln  = idx & 31;
    int j   = jt * 16 + (ln & 15);
    float v0 = 0.0f, v1 = 0.0f;
    if (ln < 16) {
      // row 0 (x<0): e = 1023-j ; row 1 (x>=0): e = j
      float e0 = floorf((float)(D_MODEL - 1 - j) * 0.5f);
      float e1 = floorf((float)j * 0.5f);
      v0 = INV_2PI * exp2f(e0 * escale);  // (1/2pi) * RES^(-2*floor(e/2)/d)
      v1 = INV_2PI * exp2f(e1 * escale);
    }
    ldsB[idx * 2 + 0] = v0;
    ldsB[idx * 2 + 1] = v1;
  }
  __syncthreads();

  // ---- Build the A operand: masked |x| outer-product column ----
  const int group = blockIdx.x * 8 + wave;   // 16-position group id
  const int nbase = group * 16;

  // branchless: lanes 16-31 re-read a valid element, then get masked to 0
  float xv = x[nbase + (lane & 15)];
  float xc = fminf(fmaxf(xv, -RES_CLAMP), RES_CLAMP);
  float ax = fabsf(xc);
  bool lo16 = (lane < 16);
  v2f a;
  a.x = (lo16 && (xc <  0.0f)) ? ax : 0.0f;  // K=0 -> row-0 table
  a.y = (lo16 && (xc >= 0.0f)) ? ax : 0.0f;  // K=1 -> row-1 table

  // ---- C matrix: per-column sin/cos phase offset (column parity = lane&1) ----
  const float off = (lane & 1) ? 0.25f : 0.0f;  // +0.25 turn == +pi/2 -> cos
  v8f c;
#pragma unroll
  for (int r = 0; r < 8; ++r) c[r] = off;

  // D VGPR r: lanes 0-15 -> row nbase+r, lanes 16-31 -> row nbase+r+8, col = jbase+(lane&15)
  float* op = out + (size_t)(nbase + ((lane >> 4) << 3)) * D_MODEL + (lane & 15);

  for (int jt = 0; jt < 64; ++jt) {
    v2f b = *(const v2f*)&ldsB[(jt * 32 + lane) * 2];  // ds_load_b64
    // t[n,j] = ax[n]*invden_scaled[row(n),j] + phase(j)  (exact, masked outer product)
    v8f t = __builtin_amdgcn_wmma_f32_16x16x4_f32(
        false, a, false, b, (short)0, c, false, false);
    float* p = op + jt * 16;
#pragma unroll
    for (int r = 0; r < 8; ++r) {
      // v_sin_f32: sin(2*pi*t) == sin(ang) (even j) / cos(ang) (odd j)
      __builtin_nontemporal_store(__builtin_amdgcn_sinf(t[r]), p + r * D_MODEL);
    }
  }
}

extern "C" void kernel_launch(void* const* d_in, const int* in_sizes, int n_in,
                              void* d_out, int out_size, void* d_ws, size_t ws_size,
                              hipStream_t stream) {
  const float* x = (const float*)d_in[0];
  float* out = (float*)d_out;
  int total = in_sizes[0];            // 2048*32 = 65536 positions
  int blocks = total / 128;           // 128 positions per block (8 waves x 16)
  posenc_angle_wmma_kernel<<<blocks, 256, 0, stream>>>(x, out);
}